// DifferentiableSimulator_87093346828580
// MI455X (gfx1250) — compile-verified
//
#include <hip/hip_runtime.h>

// ---------------------------------------------------------------------------
// Blocked linear-scan state-space simulator for MI455X (gfx1250).
//   x_{t+1} = (I+A) x_t + B u_t ;  y_t = C x_t + D u_t
// Two-level scan: 256 chunks of S=ceil(T/256) steps.
//   K1: per-chunk zero-state contributions (16 chunks/WG -> N=16 WMMA GEMMs)
//   K2: Ms = (I+A)^S via square-and-multiply WMMA GEMMs
//   K3: sequential chunk-boundary recurrence (registers-resident matrix)
//   K4: per-chunk expansion + y, emit tsol/xsol/ysol
// All matmuls in fp32 via V_WMMA_F32_16X16X4_F32 (precision-preserving; this
// problem is latency/memory bound, nowhere near the matrix-core ceiling).
// ---------------------------------------------------------------------------

typedef __attribute__((ext_vector_type(2))) float v2f;
typedef __attribute__((ext_vector_type(8))) float v8f;

#define NCHUNK 256   // number of chunks (= sequential steps in K3)

__device__ __forceinline__ v8f wmma4(v2f a, v2f b, v8f c) {
  // D = A(16x4) * B(4x16) + C(16x16), all fp32
  return __builtin_amdgcn_wmma_f32_16x16x4_f32(
      /*neg_a=*/false, a, /*neg_b=*/false, b,
      /*c_mod=*/(short)0, c, /*reuse_a=*/false, /*reuse_b=*/false);
}

// A-operand (16x4 f32): lanes 0-15 hold K=k0,k0+1; lanes 16-31 hold K=k0+2,k0+3
__device__ __forceinline__ v2f ld_a(const float* A, int lda, int m0, int k0, int lane) {
  int m = m0 + (lane & 15);
  int k = k0 + ((lane >> 4) << 1);
  const float* p = A + m * lda + k;
  v2f r; r[0] = p[0]; r[1] = p[1];
  return r;
}

// B-operand (4x16 f32): mirrors A layout (rows striped across lanes per VGPR)
__device__ __forceinline__ v2f ld_b(const float* B, int ldb, int k0, int n0, int lane) {
  int k = k0 + ((lane >> 4) << 1);
  int n = n0 + (lane & 15);
  v2f r; r[0] = B[k * ldb + n]; r[1] = B[(k + 1) * ldb + n];
  return r;
}

// ---------------------------------------------------------------------------
// M = I + A  (dt = 1)
__global__ void makeM_kernel(float* __restrict__ M, const float* __restrict__ A) {
  int i = blockIdx.x, j = threadIdx.x;
  M[i * 256 + j] = A[i * 256 + j] + (i == j ? 1.0f : 0.0f);
}

// ---------------------------------------------------------------------------
// D(256x256) = A(256x256) * B(256x256), one 16x16 tile per wave
__global__ void __launch_bounds__(256) gemm256_kernel(
    float* __restrict__ D, const float* __restrict__ A, const float* __restrict__ B) {
  const int wave = threadIdx.x >> 5, lane = threadIdx.x & 31;
  const int tile = blockIdx.x * 8 + wave;
  const int tm = (tile >> 4) * 16, tn = (tile & 15) * 16;
  v8f acc = {};
#pragma unroll 4
  for (int k0 = 0; k0 < 256; k0 += 4)
    acc = wmma4(ld_a(A, 256, tm, k0, lane), ld_b(B, 256, k0, tn, lane), acc);
  const int n = tn + (lane & 15);
  const int mb = tm + ((lane >> 4) << 3);
#pragma unroll
  for (int r = 0; r < 8; ++r) D[(mb + r) * 256 + n] = acc[r];
}

// ---------------------------------------------------------------------------
// K1: each block owns 16 chunks; state matrix X (256x16) in LDS.
// Per step: X <- X + A*X + B*U_j ; finally Vloc[c][:] = chunk-final state.
__global__ void __launch_bounds__(512) chunk_local_kernel(
    const float* __restrict__ A, const float* __restrict__ B,
    const float* __restrict__ u, float* __restrict__ Vloc, int S, int T) {
  __shared__ float X0[256 * 16], X1[256 * 16], Us[128 * 16];
  const int tid = threadIdx.x, lane = tid & 31, wave = tid >> 5;
  const int cbase = blockIdx.x * 16;
  float* Xc = X0; float* Xn = X1;
  for (int e = tid; e < 256 * 16; e += 512) X0[e] = 0.0f;
  __syncthreads();
  for (int j = 0; j < S; ++j) {
    // stage U_j for the 16 chunks: Us[k][n] = u[t(n)][k]
    for (int e = tid; e < 128 * 16; e += 512) {
      int n = e & 15, k = e >> 4;
      int t = (cbase + n) * S + j;
      if (t >= T) t = T - 1;  // pad region; only pollutes dead columns
      Us[k * 16 + n] = u[t * 128 + k];
    }
    __syncthreads();
    const int m0 = wave * 16;  // 16 waves <-> 16 row tiles of X
    v8f acc = {};
#pragma unroll 4
    for (int k0 = 0; k0 < 256; k0 += 4)
      acc = wmma4(ld_a(A, 256, m0, k0, lane), ld_b(Xc, 16, k0, 0, lane), acc);
#pragma unroll 4
    for (int k0 = 0; k0 < 128; k0 += 4)
      acc = wmma4(ld_a(B, 128, m0, k0, lane), ld_b(Us, 16, k0, 0, lane), acc);
    const int n = lane & 15, mb = m0 + ((lane >> 4) << 3);
#pragma unroll
    for (int r = 0; r < 8; ++r) acc[r] += Xc[(mb + r) * 16 + n];  // + identity
#pragma unroll
    for (int r = 0; r < 8; ++r) Xn[(mb + r) * 16 + n] = acc[r];
    __syncthreads();
    float* tp = Xc; Xc = Xn; Xn = tp;
  }
  for (int e = tid; e < 256 * 16; e += 512) {
    int n = e >> 8, i = e & 255;
    Vloc[(cbase + n) * 256 + i] = Xc[i * 16 + n];
  }
}

// ---------------------------------------------------------------------------
// K3: xb[0]=x_init; xb[c+1] = Ms*xb[c] + Vloc[c]. 512 threads, 2 per row,
// each half-row of Ms pinned in 32 float4 registers (no memory re-reads).
__global__ void __launch_bounds__(512) boundary_kernel(
    const float* __restrict__ Ms, const float* __restrict__ Vloc,
    const float* __restrict__ xinit, float* __restrict__ xb, int C) {
  __shared__ __align__(16) float xcur[256];
  __shared__ float part[512];
  const int tid = threadIdx.x, i = tid >> 1, h = tid & 1;
  float4 reg[32];
  const float4* row = (const float4*)(Ms + i * 256 + h * 128);
#pragma unroll
  for (int r = 0; r < 32; ++r) reg[r] = row[r];
  if (h == 0) { float v = xinit[i]; xcur[i] = v; xb[i] = v; }
  __syncthreads();
  for (int c = 0; c < C - 1; ++c) {
    const float4* x4p = (const float4*)(xcur + h * 128);
    float s = 0.0f;
#pragma unroll
    for (int r = 0; r < 32; ++r) {
      float4 x4 = x4p[r], m4 = reg[r];
      s += m4.x * x4.x + m4.y * x4.y + m4.z * x4.z + m4.w * x4.w;
    }
    part[tid] = s;
    __syncthreads();
    if (h == 0) {
      float v = part[2 * i] + part[2 * i + 1] + Vloc[c * 256 + i];
      xcur[i] = v;
      xb[(c + 1) * 256 + i] = v;
    }
    __syncthreads();
  }
}

// ---------------------------------------------------------------------------
// K4: expand each chunk from its true boundary state; waves 0-15 update X,
// waves 16-23 compute Y = C*X + D*U; emit tsol/xsol (pre-update x) /ysol.
__global__ void __launch_bounds__(768) emit_kernel(
    const float* __restrict__ A, const float* __restrict__ B,
    const float* __restrict__ Cm, const float* __restrict__ Dm,
    const float* __restrict__ u, const float* __restrict__ xb,
    float* __restrict__ tsol, float* __restrict__ xsol, float* __restrict__ ysol,
    int S, int T) {
  __shared__ float X0[256 * 16], X1[256 * 16], Us[128 * 16], Ys[128 * 16];
  const int tid = threadIdx.x, lane = tid & 31, wave = tid >> 5;
  const int cbase = blockIdx.x * 16;
  float* Xc = X0; float* Xn = X1;
  for (int e = tid; e < 256 * 16; e += 768) {
    int n = e >> 8, i = e & 255;
    X0[i * 16 + n] = xb[(cbase + n) * 256 + i];
  }
  __syncthreads();
  for (int j = 0; j < S; ++j) {
    for (int e = tid; e < 128 * 16; e += 768) {
      int n = e & 15, k = e >> 4;
      int t = (cbase + n) * S + j;
      if (t >= T) t = T - 1;
      Us[k * 16 + n] = u[t * 128 + k];
    }
    __syncthreads();
    if (wave < 16) {  // X-update tiles
      const int m0 = wave * 16;
      v8f acc = {};
#pragma unroll 4
      for (int k0 = 0; k0 < 256; k0 += 4)
        acc = wmma4(ld_a(A, 256, m0, k0, lane), ld_b(Xc, 16, k0, 0, lane), acc);
#pragma unroll 4
      for (int k0 = 0; k0 < 128; k0 += 4)
        acc = wmma4(ld_a(B, 128, m0, k0, lane), ld_b(Us, 16, k0, 0, lane), acc);
      const int n = lane & 15, mb = m0 + ((lane >> 4) << 3);
#pragma unroll
      for (int r = 0; r < 8; ++r) acc[r] += Xc[(mb + r) * 16 + n];
#pragma unroll
      for (int r = 0; r < 8; ++r) Xn[(mb + r) * 16 + n] = acc[r];
    } else {          // Y tiles (128x16)
      const int m0 = (wave - 16) * 16;
      v8f acc = {};
#pragma unroll 4
      for (int k0 = 0; k0 < 256; k0 += 4)
        acc = wmma4(ld_a(Cm, 256, m0, k0, lane), ld_b(Xc, 16, k0, 0, lane), acc);
#pragma unroll 4
      for (int k0 = 0; k0 < 128; k0 += 4)
        acc = wmma4(ld_a(Dm, 128, m0, k0, lane), ld_b(Us, 16, k0, 0, lane), acc);
      const int n = lane & 15, mb = m0 + ((lane >> 4) << 3);
#pragma unroll
      for (int r = 0; r < 8; ++r) Ys[(mb + r) * 16 + n] = acc[r];
    }
    __syncthreads();
    // emit (coalesced rows per chunk)
    for (int e = tid; e < 256 * 16; e += 768) {
      int n = e >> 8, i = e & 255;
      int t = (cbase + n) * S + j;
      if (t < T) {
        xsol[t * 256 + i] = Xc[i * 16 + n];
        if (i == 0) tsol[t] = (float)t;
      }
    }
    for (int e = tid; e < 128 * 16; e += 768) {
      int n = e >> 7, i = e & 127;
      int t = (cbase + n) * S + j;
      if (t < T) ysol[t * 128 + i] = Ys[i * 16 + n];
    }
    __syncthreads();
    float* tp = Xc; Xc = Xn; Xn = tp;
  }
}

// ---------------------------------------------------------------------------
extern "C" void kernel_launch(void* const* d_in, const int* in_sizes, int n_in,
                              void* d_out, int out_size, void* d_ws, size_t ws_size,
                              hipStream_t stream) {
  (void)n_in; (void)out_size; (void)ws_size;
  const float* x_init = (const float*)d_in[0];
  const float* u      = (const float*)d_in[1];
  const float* A      = (const float*)d_in[2];
  const float* B      = (const float*)d_in[3];
  const float* Cm     = (const float*)d_in[4];
  const float* Dm     = (const float*)d_in[5];
  const int T = in_sizes[1] / 128;
  const int S = (T + NCHUNK - 1) / NCHUNK;

  float* ws   = (float*)d_ws;             // layout (fp32): 7 * 64K = 1.75 MB
  float* Mv   = ws + 0 * 65536;
  float* Pa   = ws + 1 * 65536;
  float* Pb   = ws + 2 * 65536;
  float* Ra   = ws + 3 * 65536;
  float* Rb   = ws + 4 * 65536;
  float* Vloc = ws + 5 * 65536;           // NCHUNK x 256
  float* xb   = ws + 6 * 65536;           // NCHUNK x 256

  float* tsol = (float*)d_out;
  float* xsol = tsol + T;
  float* ysol = xsol + (size_t)T * 256;

  // Phase 1: chunk-local contributions (parallel over 256 chunks, 16/WG).
  chunk_local_kernel<<<NCHUNK / 16, 512, 0, stream>>>(A, B, u, Vloc, S, T);

  // Phase 2a: Ms = (I+A)^S via square-and-multiply (~12 WMMA GEMMs).
  makeM_kernel<<<256, 256, 0, stream>>>(Mv, A);
  const float* P = Mv; const float* R = nullptr;
  float* Psq[2] = {Pa, Pb}; int pi = 0;
  float* Rbf[2] = {Ra, Rb}; int ri = 0;
  unsigned e = (unsigned)S;
  while (true) {
    if (e & 1u) {
      if (!R) {
        if (P == Mv) { R = Mv; }
        else { hipMemcpyAsync(Rbf[ri], P, 65536 * sizeof(float),
                              hipMemcpyDeviceToDevice, stream); R = Rbf[ri]; ri ^= 1; }
      } else {
        gemm256_kernel<<<32, 256, 0, stream>>>(Rbf[ri], R, P); R = Rbf[ri]; ri ^= 1;
      }
    }
    e >>= 1;
    if (!e) break;
    gemm256_kernel<<<32, 256, 0, stream>>>(Psq[pi], P, P); P = Psq[pi]; pi ^= 1;
  }

  // Phase 2b: sequential boundary recurrence (NCHUNK dependent matvecs).
  boundary_kernel<<<1, 512, 0, stream>>>(R, Vloc, x_init, xb, NCHUNK);

  // Phase 3: expand chunks, compute y, emit all outputs.
  emit_kernel<<<NCHUNK / 16, 768, 0, stream>>>(A, B, Cm, Dm, u, xb,
                                               tsol, xsol, ysol, S, T);
}